// Model_39676907882853
// MI455X (gfx1250) — compile-verified
//
#include <hip/hip_runtime.h>
#include <hip/hip_bf16.h>
#include <math.h>

// Problem constants (match reference)
#define Bsz 4
#define Nseq 1024
#define Dmodel 1024
#define Hh 16
#define DH 64
#define Zz (Bsz * Hh)         // 64
#define THREE_D (3 * Dmodel)  // 3072

typedef float v2f __attribute__((ext_vector_type(2)));
typedef float v4f __attribute__((ext_vector_type(4)));
typedef float v8f __attribute__((ext_vector_type(8)));

// ---------------------------------------------------------------------------
// Kernel 1: qkv = query @ W_qkv^T + b_qkv, scattered into head-split q/k/v.
//   K is stored TRANSPOSED ([z][dh][n]) so the attention kernel's B-operand
//   reads are coalesced.
//   Block: 256 threads (8 waves), tile 32(M) x 64(N), K staged in LDS (KT=32)
// ---------------------------------------------------------------------------
#define ASTR 36   // 32+4: rows 16B-aligned (36*4=144) and conflict-free
__global__ __launch_bounds__(256) void qkv_gemm_kernel(
    const float* __restrict__ A,     // query [4096][1024]
    const float* __restrict__ W,     // W_qkv [3072][1024]
    const float* __restrict__ bias,  // [3072]
    float* __restrict__ qws,         // [64][1024][64]
    float* __restrict__ kws,         // [64][64][1024]  (transposed)
    float* __restrict__ vws) {       // [64][1024][64]
  __shared__ float As[32 * ASTR];
  __shared__ float Bs[64 * ASTR];

  const int tid  = threadIdx.x;
  const int w    = tid >> 5;
  const int lane = tid & 31;
  const int wm   = w & 1;        // M subtile (0..1)
  const int wn   = w >> 1;       // N subtile (0..3)
  const int m0   = blockIdx.x * 32;
  const int n0   = blockIdx.y * 64;

  const int lane15 = lane & 15;
  const int k0     = (lane >> 4) * 2;   // f32 WMMA K-pair select per lane half
  const int arow   = wm * 16 + lane15;
  const int bcol   = wn * 16 + lane15;

  // float4 staging coordinates
  const int ar4 = tid >> 3;            // 0..31
  const int ak4 = (tid & 7) * 4;       // 0,4,..,28

  v8f acc = {0.f, 0.f, 0.f, 0.f, 0.f, 0.f, 0.f, 0.f};

  for (int kb = 0; kb < Dmodel; kb += 32) {
    // Stage A tile 32x32: one b128 load/store per thread
    *(v4f*)&As[ar4 * ASTR + ak4] =
        *(const v4f*)(A + (size_t)(m0 + ar4) * Dmodel + kb + ak4);
    // Stage B tile 64x32: two b128 per thread
    for (int i = 0; i < 2; ++i) {
      int idx = i * 256 + tid;
      int r = idx >> 3, kk4 = (idx & 7) * 4;
      *(v4f*)&Bs[r * ASTR + kk4] =
          *(const v4f*)(W + (size_t)(n0 + r) * Dmodel + kb + kk4);
    }
    __syncthreads();

    for (int kk = 0; kk < 32; kk += 4) {
      v2f a, b;
      a.x = As[arow * ASTR + kk + k0];
      a.y = As[arow * ASTR + kk + k0 + 1];
      b.x = Bs[bcol * ASTR + kk + k0];
      b.y = Bs[bcol * ASTR + kk + k0 + 1];
      acc = __builtin_amdgcn_wmma_f32_16x16x4_f32(
          false, a, false, b, (short)0, acc, false, false);
    }
    __syncthreads();
  }

  // Store: fuse bias add + split into q/k/v layouts
  const int col   = n0 + wn * 16 + lane15;
  const float bv  = bias[col];
  const int which = col >> 10;     // 0=q 1=k 2=v
  const int f     = col & 1023;
  const int h     = f >> 6;
  const int dd    = f & 63;

  for (int r = 0; r < 8; ++r) {
    int row = wm * 16 + ((lane < 16) ? r : r + 8);
    int m   = m0 + row;
    int bb  = m >> 10;          // batch
    int n   = m & 1023;         // seq pos
    int zz  = bb * Hh + h;
    float val = acc[r] + bv;
    if (which == 0) {
      qws[(size_t)zz * (Nseq * DH) + (size_t)n * DH + dd] = val;
    } else if (which == 1) {
      // transposed: [z][dh][n]
      kws[(size_t)zz * (Nseq * DH) + (size_t)dd * Nseq + n] = val;
    } else {
      vws[(size_t)zz * (Nseq * DH) + (size_t)n * DH + dd] = val;
    }
  }
}

// ---------------------------------------------------------------------------
// Kernel 2: attention for one (z, 16-row block).
//   S = (Q Kt) * 8 ; P = softmax(S) * mask / Z ; O = P V ; fused merge-heads
//   Block: 256 threads (8 waves).
// ---------------------------------------------------------------------------
#define SSTR 1028   // 1028*4 % 16 == 0 (aligned b128) ; rows on distinct banks
#define QSTR 68     // 68*4 = 272, 16B aligned ; 68%64=4 -> conflict-free rows

__global__ __launch_bounds__(256) void attn_kernel(
    const float* __restrict__ qws,   // [64][1024][64]
    const float* __restrict__ kwsT,  // [64][64][1024] (dh-major)
    const float* __restrict__ vws,   // [64][1024][64]
    const float* __restrict__ mask,  // [64][1024][1024]
    float* __restrict__ out) {       // [4][1024][1024]
  __shared__ float Qs[16 * QSTR];
  __shared__ float Ss[16 * SSTR];
  __shared__ float red[16 * 17];
  __shared__ float rowmax[16];
  __shared__ float rowsum[16];
  __shared__ float Opart[8 * 256];

  const int z      = blockIdx.x;         // head-batch 0..63
  const int rowblk = blockIdx.y;         // 0..63 (16 rows each)
  const int tid    = threadIdx.x;
  const int w      = tid >> 5;
  const int lane   = tid & 31;
  const int lane15 = lane & 15;
  const int k0     = (lane >> 4) * 2;

  const size_t zoff = (size_t)z * (Nseq * DH);
  const int row0    = rowblk * 16;

  // ---- Load Q tile (16 x 64) into LDS: one b128 per thread ----
  {
    int r = tid >> 4, d4 = (tid & 15) * 4;
    *(v4f*)&Qs[r * QSTR + d4] =
        *(const v4f*)(qws + zoff + (size_t)(row0 + r) * DH + d4);
  }
  __syncthreads();

  // ---- Phase B: S = Q @ K^T, wave w owns columns [w*128, w*128+128) ----
  const float* KT = kwsT + zoff;   // [64][1024] dh-major
  for (int ct = 0; ct < 8; ++ct) {
    const int colbase = w * 128 + ct * 16;
    const int keycol  = colbase + lane15;
    v8f acc = {0.f, 0.f, 0.f, 0.f, 0.f, 0.f, 0.f, 0.f};
    for (int d = 0; d < DH; d += 4) {
      v2f a, b;
      a.x = Qs[lane15 * QSTR + d + k0];
      a.y = Qs[lane15 * QSTR + d + k0 + 1];
      b.x = KT[(size_t)(d + k0) * Nseq + keycol];      // coalesced across lanes
      b.y = KT[(size_t)(d + k0 + 1) * Nseq + keycol];
      acc = __builtin_amdgcn_wmma_f32_16x16x4_f32(
          false, a, false, b, (short)0, acc, false, false);
    }
    // quirk scale: qk * sqrt(DH) = *8
    for (int r = 0; r < 8; ++r) {
      int row = (lane < 16) ? r : r + 8;
      Ss[row * SSTR + colbase + lane15] = acc[r] * 8.0f;
    }
  }
  __syncthreads();

  // ---- Phase C: row softmax + dropout mask, 4-wide in LDS ----
  const int srow = tid >> 4;        // 0..15
  const int scol = tid & 15;        // 0..15
  const int cbase = scol * 4;       // thread owns cols cbase+64*i .. +3
  float m = -INFINITY;
  for (int i = 0; i < 16; ++i) {
    v4f s4 = *(const v4f*)&Ss[srow * SSTR + cbase + i * 64];
    m = fmaxf(m, fmaxf(fmaxf(s4.x, s4.y), fmaxf(s4.z, s4.w)));
  }
  red[srow * 17 + scol] = m;
  __syncthreads();
  if (scol == 0) {
    float mm = red[srow * 17];
    for (int j = 1; j < 16; ++j) mm = fmaxf(mm, red[srow * 17 + j]);
    rowmax[srow] = mm;
  }
  __syncthreads();
  const float mx = rowmax[srow];
  float s = 0.f;
  for (int i = 0; i < 16; ++i) {
    v4f s4 = *(v4f*)&Ss[srow * SSTR + cbase + i * 64];
    v4f p;
    p.x = __expf(s4.x - mx);
    p.y = __expf(s4.y - mx);
    p.z = __expf(s4.z - mx);
    p.w = __expf(s4.w - mx);
    *(v4f*)&Ss[srow * SSTR + cbase + i * 64] = p;
    s += p.x + p.y + p.z + p.w;
  }
  red[srow * 17 + scol] = s;
  __syncthreads();
  if (scol == 0) {
    float ss = 0.f;
    for (int j = 0; j < 16; ++j) ss += red[srow * 17 + j];
    rowsum[srow] = ss;
  }
  __syncthreads();
  const float inv = 1.0f / rowsum[srow];
  const float* mrow =
      mask + (size_t)z * Nseq * Nseq + (size_t)(row0 + srow) * Nseq;
  for (int i = 0; i < 16; ++i) {
    int col = cbase + i * 64;
    v4f mv = *(const v4f*)(mrow + col);     // b128, coalesced mask read
    v4f p  = *(v4f*)&Ss[srow * SSTR + col];
    p.x *= mv.x * inv;
    p.y *= mv.y * inv;
    p.z *= mv.z * inv;
    p.w *= mv.w * inv;
    *(v4f*)&Ss[srow * SSTR + col] = p;      // inverted dropout + 1/Z
  }
  __syncthreads();

  // ---- Phase D: O = P @ V, split-K across wave pairs ----
  const int nt   = w & 3;        // dh column tile (0..3)
  const int kh   = w >> 2;       // K half (0..1)
  const int kbeg = kh * 512;
  const int vcol = nt * 16 + lane15;
  const float* Vg = vws + zoff;
  {
    v8f acc = {0.f, 0.f, 0.f, 0.f, 0.f, 0.f, 0.f, 0.f};
    for (int kk = kbeg; kk < kbeg + 512; kk += 4) {
      v2f a, b;
      a.x = Ss[lane15 * SSTR + kk + k0];
      a.y = Ss[lane15 * SSTR + kk + k0 + 1];
      b.x = Vg[(size_t)(kk + k0) * DH + vcol];
      b.y = Vg[(size_t)(kk + k0 + 1) * DH + vcol];
      acc = __builtin_amdgcn_wmma_f32_16x16x4_f32(
          false, a, false, b, (short)0, acc, false, false);
    }
    for (int r = 0; r < 8; ++r) {
      int row = (lane < 16) ? r : r + 8;
      Opart[w * 256 + row * 16 + lane15] = acc[r];
    }
  }
  __syncthreads();

  // ---- Reduce split-K partials + fused merge-heads b128 store ----
  const int bb = z >> 4;   // batch
  const int hh = z & 15;   // head
  {
    int r = tid >> 4;                // 0..15
    int dcol = (tid & 15) * 4;       // 0,4,..,60 (stays inside one 16-tile)
    int t = dcol >> 4, cc = dcol & 15;
    v4f a = *(const v4f*)&Opart[t * 256 + r * 16 + cc];
    v4f b = *(const v4f*)&Opart[(t + 4) * 256 + r * 16 + cc];
    v4f val;
    val.x = a.x + b.x;
    val.y = a.y + b.y;
    val.z = a.z + b.z;
    val.w = a.w + b.w;
    *(v4f*)(out + ((size_t)bb * Nseq + (row0 + r)) * Dmodel + hh * DH + dcol) =
        val;
  }
}

// ---------------------------------------------------------------------------
extern "C" void kernel_launch(void* const* d_in, const int* in_sizes, int n_in,
                              void* d_out, int out_size, void* d_ws, size_t ws_size,
                              hipStream_t stream) {
  (void)in_sizes; (void)n_in; (void)out_size; (void)ws_size;
  const float* query = (const float*)d_in[0];
  // d_in[1] (key) and d_in[2] (value) are unused by the reference module
  const float* W_qkv = (const float*)d_in[3];
  const float* b_qkv = (const float*)d_in[4];
  const float* dmask = (const float*)d_in[5];
  float* out = (float*)d_out;

  // Workspace: q/v head-split [64][1024][64], k transposed [64][64][1024]
  const size_t per = (size_t)Zz * Nseq * DH;   // 4,194,304 floats
  float* qws = (float*)d_ws;
  float* kws = qws + per;
  float* vws = kws + per;

  dim3 g1((Bsz * Nseq) / 32, THREE_D / 64);   // (128, 48)
  qkv_gemm_kernel<<<g1, 256, 0, stream>>>(query, W_qkv, b_qkv, qws, kws, vws);

  dim3 g2(Zz, Nseq / 16);                     // (64, 64)
  attn_kernel<<<g2, 256, 0, stream>>>(qws, kws, vws, dmask, out);
}